// QSAL_pennylane_75170517614993
// MI455X (gfx1250) — compile-verified
//
#include <hip/hip_runtime.h>
#include <math.h>

// ---------------- problem constants (match reference) ----------------
#define NQ      10
#define NSTATE  1024      // 2^10 amplitudes
#define SEQ     16
#define BATCH   32
#define DFEAT   60        // per-position feature / weight length
#define THREADS 256       // 8 wave32 per circuit-sim block

typedef __attribute__((ext_vector_type(2))) float v2f;
typedef __attribute__((ext_vector_type(8))) float v8f;

// Wire j of the reference maps to flat-index bit (9 - j)  (axis 0 = MSB).

// Apply RX/RY(theta) on `bit`; c = cos(theta/2), s = sin(theta/2).
// 512 pairs, 2 per thread. Barrier at end.
__device__ __forceinline__ void rot_gate(float* re, float* im, int bit,
                                         float c, float s, int isRX, int tid) {
  const int mask = 1 << bit;
  #pragma unroll
  for (int p = tid; p < (NSTATE / 2); p += THREADS) {
    const int i0 = ((p & ~(mask - 1)) << 1) | (p & (mask - 1));
    const int i1 = i0 | mask;
    const float r0 = re[i0], q0 = im[i0];
    const float r1 = re[i1], q1 = im[i1];
    float nr0, nq0, nr1, nq1;
    if (isRX) {           // [[c, -i s], [-i s, c]]
      nr0 = c * r0 + s * q1;  nq0 = c * q0 - s * r1;
      nr1 = c * r1 + s * q0;  nq1 = c * q1 - s * r0;
    } else {              // [[c, -s], [s, c]]
      nr0 = c * r0 - s * r1;  nq0 = c * q0 - s * q1;
      nr1 = s * r0 + c * r1;  nq1 = s * q0 + c * q1;
    }
    re[i0] = nr0; im[i0] = nq0;
    re[i1] = nr1; im[i1] = nq1;
  }
  __syncthreads();
}

// CNOT(control bit bc -> target bit bt): swap amp[n] <-> amp[n|tm] for
// all n with control=1, target=0. 256 such pairs = one per thread.
__device__ __forceinline__ void cnot_gate(float* re, float* im,
                                          int bc, int bt, int tid) {
  const int cm = 1 << bc, tm = 1 << bt;
  const int lo = (bc < bt) ? bc : bt;
  const int hi = (bc < bt) ? bt : bc;
  int n = tid;                                        // 8 free bits
  n = ((n & ~((1 << lo) - 1)) << 1) | (n & ((1 << lo) - 1));  // insert 0 @ lo
  n = ((n & ~((1 << hi) - 1)) << 1) | (n & ((1 << hi) - 1));  // insert 0 @ hi
  n |= cm;                                            // control = 1, target = 0
  const int n2 = n | tm;
  const float tr = re[n], ti = im[n];
  re[n] = re[n2];  im[n] = im[n2];
  re[n2] = tr;     im[n2] = ti;
  __syncthreads();
}

// One block = one circuit simulation. grid = (512, 3): x = b*16+s, y = Q/K/V.
__global__ __launch_bounds__(THREADS)
void qsal_sim_kernel(const float* __restrict__ x,
                     const float* __restrict__ Wq,
                     const float* __restrict__ Wk,
                     const float* __restrict__ Wv,
                     const int*   __restrict__ pidx,
                     float*       __restrict__ ws_qkv) {
  __shared__ float re[NSTATE];
  __shared__ float im[NSTATE];
  __shared__ float cA[2 * DFEAT];   // cos(theta/2): [0,60)=enc(x), [60,120)=var(W)
  __shared__ float sA[2 * DFEAT];
  __shared__ float red[THREADS];

  const int tid   = threadIdx.x;
  const int bs    = blockIdx.x;       // b*16 + s
  const int which = blockIdx.y;       // 0=Q, 1=K, 2=V
  const int s     = bs & 15;
  const float* W  = (which == 0) ? Wq : (which == 1) ? Wk : Wv;

  // Precompute half-angle cos/sin for all 120 rotation angles.
  for (int t = tid; t < 2 * DFEAT; t += THREADS) {
    const float a = (t < DFEAT) ? x[bs * DFEAT + t] : W[s * DFEAT + (t - DFEAT)];
    cA[t] = cosf(0.5f * a);
    sA[t] = sinf(0.5f * a);
  }
  // |0...0>
  for (int n = tid; n < NSTATE; n += THREADS) {
    re[n] = (n == 0) ? 1.0f : 0.0f;
    im[n] = 0.0f;
  }
  __syncthreads();

  // Encoding block (angles from x), then variational block (angles from W);
  // identical structure.
  for (int blk = 0; blk < 2; ++blk) {
    const int base = blk * DFEAT;
    int idx = 0;
    for (int j = 0; j < NQ; ++j) {
      rot_gate(re, im, 9 - j, cA[base + idx],     sA[base + idx],     1, tid); // RX
      rot_gate(re, im, 9 - j, cA[base + idx + 1], sA[base + idx + 1], 0, tid); // RY
      idx += 2;
    }
    for (int l = 0; l < 4; ++l) {
      for (int j = 0; j < NQ; ++j)
        cnot_gate(re, im, 9 - j, 9 - ((j + 1) % NQ), tid);
      for (int j = 0; j < NQ; ++j) {
        rot_gate(re, im, 9 - j, cA[base + idx], sA[base + idx], 0, tid);       // RY
        ++idx;
      }
    }
  }

  // ------------- measurement -------------
  float local = 0.0f;
  if (which < 2) {
    // <Z on wire 0> = bit 9: sum (+|a|^2 if bit9==0 else -|a|^2)
    for (int n = tid; n < NSTATE; n += THREADS) {
      const float a = re[n], b = im[n];
      const float p = a * a + b * b;
      local += (n & 512) ? -p : p;
    }
  } else {
    // <psi| P |psi>, P = tensor Pauli string: sum_n conj(psi[n]) * phi(n^xm) * psi[n^xm]
    int xm = 0, ym = 0, zm = 0;
    for (int j = 0; j < NQ; ++j) {
      const int c = pidx[j];
      const int bm = 1 << (9 - j);
      if (c == 1)       { xm |= bm; }
      else if (c == 2)  { xm |= bm; ym |= bm; }
      else if (c == 3)  { zm |= bm; }
    }
    const int nY = __popc(ym) & 3;                 // global i^{nY}
    const float pr  = (nY == 0) ? 1.0f : (nY == 2) ? -1.0f : 0.0f;
    const float pii = (nY == 1) ? 1.0f : (nY == 3) ? -1.0f : 0.0f;
    const int sm = ym | zm;                        // sign bits
    for (int n = tid; n < NSTATE; n += THREADS) {
      const int m = n ^ xm;
      const float sgn = (__popc(m & sm) & 1) ? -1.0f : 1.0f;
      const float ar = re[n], ai = im[n];
      const float br = re[m], bi = im[m];
      const float X = ar * br + ai * bi;           // Re(conj(a)*b)
      const float Y = ar * bi - ai * br;           // Im(conj(a)*b)
      local += sgn * (pr * X - pii * Y);
    }
  }

  red[tid] = local;
  __syncthreads();
  for (int off = THREADS / 2; off > 0; off >>= 1) {
    if (tid < off) red[tid] += red[tid + off];
    __syncthreads();
  }
  if (tid == 0) ws_qkv[which * (BATCH * SEQ) + bs] = red[0];
}

// One wave32 per batch. G[i][j] = exp(-(Q[i]-K[j])^2) (= alpha^T).
// num = G @ Vexp (col 0), den = G^T @ ones (col 0), via chained
// V_WMMA_F32_16X16X4_F32 (full fp32 -> same numerics as FMA path).
__global__ __launch_bounds__(32)
void qsal_attn_kernel(const float* __restrict__ x,
                      const float* __restrict__ qkv,
                      float*       __restrict__ out) {
  const int b    = blockIdx.x;        // 0..31
  const int lane = threadIdx.x;       // 0..31
  const int half = lane >> 4;         // A/B frag: lanes 16-31 hold K+2
  const int nm   = lane & 15;         // row (A) / column (B,D) index

  const float* Qp = qkv + 0 * (BATCH * SEQ) + b * SEQ;
  const float* Kp = qkv + 1 * (BATCH * SEQ) + b * SEQ;
  const float* Vp = qkv + 2 * (BATCH * SEQ) + b * SEQ;

  const float qm = Qp[nm];            // Q[row] for A1
  const float km = Kp[nm];            // K[row] for A2 (G^T)

  v8f acc1 = {};                      // D1 = G   @ [Vexp | 0...]
  v8f acc2 = {};                      // D2 = G^T @ [1    | 0...]

  #pragma unroll
  for (int c = 0; c < 4; ++c) {
    const int k0 = 4 * c + 2 * half;  // this lane's two K indices
    const float K0 = Kp[k0], K1 = Kp[k0 + 1];
    const float Q0 = Qp[k0], Q1 = Qp[k0 + 1];

    v2f a1, a2, b1, b2;
    const float d0 = qm - K0, d1 = qm - K1;
    a1.x = expf(-d0 * d0);  a1.y = expf(-d1 * d1);        // G[row][k0], G[row][k0+1]
    const float e0 = Q0 - km, e1 = Q1 - km;
    a2.x = expf(-e0 * e0);  a2.y = expf(-e1 * e1);        // G^T[row][k]

    const float v0 = Vp[k0], v1 = Vp[k0 + 1];
    b1.x = (nm == 0) ? v0 : 0.0f;  b1.y = (nm == 0) ? v1 : 0.0f;  // col0 = Vexp
    b2.x = (nm == 0) ? 1.0f : 0.0f; b2.y = b2.x;                  // col0 = ones

    acc1 = __builtin_amdgcn_wmma_f32_16x16x4_f32(
        false, a1, false, b1, (short)0, acc1, false, false);
    acc2 = __builtin_amdgcn_wmma_f32_16x16x4_f32(
        false, a2, false, b2, (short)0, acc2, false, false);
  }

  // D layout: lane {0,16} hold column 0; VGPR v -> row v + 8*half.
  __shared__ float s_num[SEQ];
  __shared__ float s_den[SEQ];
  if (nm == 0) {
    #pragma unroll
    for (int v = 0; v < 8; ++v) {
      s_num[v + 8 * half] = acc1[v];
      s_den[v + 8 * half] = acc2[v];
    }
  }
  __syncthreads();

  const int baseo = b * (SEQ * DFEAT);
  for (int t = lane; t < SEQ * DFEAT; t += 32) {
    const int i = t / DFEAT;
    out[baseo + t] = x[baseo + t] + s_num[i] / s_den[i];
  }
}

extern "C" void kernel_launch(void* const* d_in, const int* in_sizes, int n_in,
                              void* d_out, int out_size, void* d_ws, size_t ws_size,
                              hipStream_t stream) {
  (void)in_sizes; (void)n_in; (void)out_size; (void)ws_size;
  const float* x    = (const float*)d_in[0];   // (32,16,60)
  const float* Wq   = (const float*)d_in[1];   // (16,60)
  const float* Wk   = (const float*)d_in[2];
  const float* Wv   = (const float*)d_in[3];
  const int*   pidx = (const int*)d_in[4];     // (10,)
  float* qkv = (float*)d_ws;                   // 3 * 512 floats: Q | K | Vexp
  float* out = (float*)d_out;

  // 1536 independent circuit simulations, one block each.
  qsal_sim_kernel<<<dim3(BATCH * SEQ, 3, 1), THREADS, 0, stream>>>(
      x, Wq, Wk, Wv, pidx, qkv);
  // Attention epilogue: one wave per batch, fp32 WMMA 16x16 tiles.
  qsal_attn_kernel<<<dim3(BATCH, 1, 1), 32, 0, stream>>>(x, qkv, out);
}